// HeteroRGCNLayer_86861418595102
// MI455X (gfx1250) — compile-verified
//
#include <hip/hip_runtime.h>

typedef __attribute__((ext_vector_type(2))) float v2f;
typedef __attribute__((ext_vector_type(8))) float v8f;

#define DIN  512
#define DOUT 512

// ---------------------------------------------------------------------------
// Zero-fill (d_out / deg are poisoned by harness)
// ---------------------------------------------------------------------------
__global__ void fill_zero_kernel(float* __restrict__ p, size_t count) {
    size_t i      = (size_t)blockIdx.x * blockDim.x + threadIdx.x;
    size_t stride = (size_t)gridDim.x * blockDim.x;
    for (; i < count; i += stride) p[i] = 0.0f;
}

// ---------------------------------------------------------------------------
// Degree count per (relation, node): deg[r*n + dst] += 1
// ---------------------------------------------------------------------------
__global__ void rgcn_degree_kernel(const int* __restrict__ dst,
                                   float* __restrict__ deg,
                                   int e_per_rel, int n, int total) {
    int i = blockIdx.x * blockDim.x + threadIdx.x;
    if (i >= total) return;
    int r = i / e_per_rel;
    atomicAdd(&deg[(size_t)r * n + dst[i]], 1.0f);
}

// ---------------------------------------------------------------------------
// WMMA f32 GEMM: Wh = x @ W + b  for one relation.
//   x : [nrows, 512] row-major, W : [512, 512] row-major, b : [512]
// Block: 128 threads (4 waves). Block tile 64(M) x 128(N); each wave owns a
// 32x64 quadrant = 2x4 accumulators of V_WMMA_F32_16X16X4_F32.
// K staged via LDS in 32-wide slabs. W tile stored TRANSPOSED ([N][K], odd
// stride) so both A and B fragments are contiguous K-pairs -> each fragment
// is one ds_load_2addr_b32 into an aligned VGPR pair, no repack movs.
// Fragment reuse: 6 LDS loads feed 8 WMMAs per k-step.
// ---------------------------------------------------------------------------
__global__ __launch_bounds__(128)
void rgcn_gemm_wmma_kernel(const float* __restrict__ x,
                           const float* __restrict__ W,
                           const float* __restrict__ b,
                           float* __restrict__ Wh,
                           int nrows) {
    __shared__ float xs[64][33];    // M x K slab (odd stride: conflict-free)
    __shared__ float wsT[128][33];  // N x K slab, transposed

    const int tid  = threadIdx.x;
    const int wave = tid >> 5;         // 0..3
    const int lane = tid & 31;
    const int l15  = lane & 15;
    const int khal = (lane >> 4) * 2;  // lanes 0-15 -> K+0/K+1, lanes 16-31 -> K+2/K+3

    const int m0 = blockIdx.x * 64;    // output row tile
    const int n0 = blockIdx.y * 128;   // output col tile
    const int wm = (wave >> 1) * 32;   // wave's M offset inside tile: 0/32
    const int wn = (wave & 1) * 64;    // wave's N offset inside tile: 0/64

    v8f acc[2][4] = {};

    for (int k0 = 0; k0 < DIN; k0 += 32) {
        // Stage x tile: 64 rows x 32 K (coalesced over K)
        for (int idx = tid; idx < 64 * 32; idx += 128) {
            int r = idx >> 5, c = idx & 31;
            int gr = m0 + r;
            if (gr >= nrows) gr = nrows - 1;      // clamp; stores are guarded
            xs[r][c] = x[(size_t)gr * DIN + k0 + c];
        }
        // Stage W tile transposed: wsT[col][k] (global read coalesced over col,
        // LDS write stride 33 dwords -> conflict-free)
        for (int idx = tid; idx < 32 * 128; idx += 128) {
            int k = idx >> 7, c = idx & 127;
            wsT[c][k] = W[(size_t)(k0 + k) * DOUT + n0 + c];
        }
        __syncthreads();

        #pragma unroll
        for (int kk = 0; kk < 32; kk += 4) {
            const int kb = kk + khal;
            // A fragments (16x4 f32): lane&15 = M row, VGPR pair = adjacent K
            v2f a[2];
            #pragma unroll
            for (int i = 0; i < 2; ++i) {
                a[i].x = xs[wm + i * 16 + l15][kb];
                a[i].y = xs[wm + i * 16 + l15][kb + 1];
            }
            // B fragments (4x16 f32): lane&15 = N col, VGPR pair = adjacent K
            v2f bf[4];
            #pragma unroll
            for (int j = 0; j < 4; ++j) {
                bf[j].x = wsT[wn + j * 16 + l15][kb];
                bf[j].y = wsT[wn + j * 16 + l15][kb + 1];
            }
            #pragma unroll
            for (int i = 0; i < 2; ++i)
                #pragma unroll
                for (int j = 0; j < 4; ++j)
                    acc[i][j] = __builtin_amdgcn_wmma_f32_16x16x4_f32(
                        false, a[i], false, bf[j], (short)0, acc[i][j], false, false);
        }
        __syncthreads();
    }

    // C layout: VGPR v -> M = v + (lane>=16 ? 8 : 0), N = lane&15
    const int mbase = m0 + wm + (lane >> 4) * 8;
    #pragma unroll
    for (int i = 0; i < 2; ++i) {
        #pragma unroll
        for (int j = 0; j < 4; ++j) {
            const int col  = n0 + wn + j * 16 + l15;
            const float bb = b[col];
            #pragma unroll
            for (int v = 0; v < 8; ++v) {
                const int row = mbase + i * 16 + v;
                if (row < nrows)
                    Wh[(size_t)row * DOUT + col] = acc[i][j][v] + bb;
            }
        }
    }
}

// ---------------------------------------------------------------------------
// Edge gather + scaled scatter-add: out[dst] += Wh[src] / max(deg[dst],1)
// 2 edges per 256-thread block; 128 threads x float4 = 512 floats per edge.
// Wh (102 MB/relation) is L2-resident -> gathers & atomics hit L2.
// ---------------------------------------------------------------------------
__global__ __launch_bounds__(256)
void rgcn_edge_agg_kernel(const float* __restrict__ Wh,
                          const int* __restrict__ src,
                          const int* __restrict__ dst,
                          const float* __restrict__ deg,
                          float* __restrict__ out,
                          int nedges) {
    const int e = blockIdx.x * 2 + (threadIdx.x >> 7);
    if (e >= nedges) return;
    const int t = threadIdx.x & 127;
    const int s = src[e];
    const int d = dst[e];
    const float inv = 1.0f / fmaxf(deg[d], 1.0f);

    const float4 v = *(const float4*)(Wh + (size_t)s * DOUT + t * 4);
    float* o = out + (size_t)d * DOUT + t * 4;
    atomicAdd(o + 0, v.x * inv);
    atomicAdd(o + 1, v.y * inv);
    atomicAdd(o + 2, v.z * inv);
    atomicAdd(o + 3, v.w * inv);
}

// ---------------------------------------------------------------------------
extern "C" void kernel_launch(void* const* d_in, const int* in_sizes, int n_in,
                              void* d_out, int out_size, void* d_ws, size_t ws_size,
                              hipStream_t stream) {
    const float* x   = (const float*)d_in[0];   // [n, 512]
    const float* W   = (const float*)d_in[1];   // [r, 512, 512]
    const float* b   = (const float*)d_in[2];   // [r, 512]
    const int*   src = (const int*)d_in[3];     // [r, e]
    const int*   dst = (const int*)d_in[4];     // [r, e]
    float*       out = (float*)d_out;           // [n, 512]

    const int n    = in_sizes[0] / DIN;                 // 50000
    const int nrel = in_sizes[1] / (DIN * DOUT);        // 3
    const int e    = in_sizes[3] / nrel;                // 150000

    // Workspace layout: Wh [n*512] f32 (one relation at a time, L2-resident),
    // then deg [nrel*n] f32.
    float* Wh  = (float*)d_ws;
    float* deg = Wh + (size_t)n * DOUT;

    // Zero output and degree arrays
    fill_zero_kernel<<<2048, 256, 0, stream>>>(out, (size_t)n * DOUT);
    fill_zero_kernel<<<512, 256, 0, stream>>>(deg, (size_t)nrel * n);

    // Degree count across all relations
    {
        const int total  = nrel * e;
        const int blocks = (total + 255) / 256;
        rgcn_degree_kernel<<<blocks, 256, 0, stream>>>(dst, deg, e, n, total);
    }

    // Per relation: WMMA GEMM into Wh, then edge aggregation into out.
    dim3 ggrid((n + 63) / 64, DOUT / 128, 1);
    for (int r = 0; r < nrel; ++r) {
        rgcn_gemm_wmma_kernel<<<ggrid, 128, 0, stream>>>(
            x, W + (size_t)r * DIN * DOUT, b + (size_t)r * DOUT, Wh, n);
        rgcn_edge_agg_kernel<<<(e + 1) / 2, 256, 0, stream>>>(
            Wh, src + (size_t)r * e, dst + (size_t)r * e,
            deg + (size_t)r * n, out, e);
    }
}